// _ECELoss_5729486372991
// MI455X (gfx1250) — compile-verified
//
#include <hip/hip_runtime.h>
#include <hip/hip_bf16.h>
#include <stdint.h>

typedef unsigned int v4u __attribute__((ext_vector_type(4)));
typedef int          v8i __attribute__((ext_vector_type(8)));
typedef int          v4i __attribute__((ext_vector_type(4)));

#define N_BINS        15
#define TILE_ROWS     64
#define C_DIM         100
#define LANES_USED    (C_DIM / 4)      // 25 lanes x float4 per row
#define BLOCK_THREADS 256
#define WAVES_PER_BLK (BLOCK_THREADS / 32)

// ---------------------------------------------------------------------------
// Issue a TDM async tensor load: [rows x 100] f32 tile, row-major, from global
// memory into LDS at byte offset `lds_byte`.  Completion tracked on TENSORcnt.
// D# layout per CDNA5 ISA ch.8 (CDNA5 ISA).
// ---------------------------------------------------------------------------
__device__ __forceinline__ void tdm_load_tile_to_lds(uint32_t lds_byte,
                                                     const float* gsrc,
                                                     int rows) {
  uint64_t ga = (uint64_t)(uintptr_t)gsrc;
  v4u g0;
  g0.x = 1u;                                   // count=1 valid descriptor, user mode
  g0.y = lds_byte;                             // lds_addr (bytes)
  g0.z = (uint32_t)ga;                         // global_addr[31:0]
  g0.w = (uint32_t)((ga >> 32) & 0x1FFFFFFull) // global_addr[56:32]
         | (2u << 30);                         // type=2 ("image")

  const uint32_t tdim0   = C_DIM;              // tensor width (elements)
  const uint32_t tdim1   = 0x7FFFFFFFu;        // rows: tiles always in-bounds
  const uint32_t stride0 = C_DIM;              // row stride (elements)

  v8i g1;
  g1[0] = (int)(2u << 16);                     // wg_mask=0, data_size=2 (4B), no pad/iter
  g1[1] = (int)((tdim0 & 0xFFFFu) << 16);      // [63:48] tensor_dim0[15:0]
  g1[2] = (int)(((tdim0 >> 16) & 0xFFFFu) | ((tdim1 & 0xFFFFu) << 16));
  g1[3] = (int)(((tdim1 >> 16) & 0xFFFFu) | ((uint32_t)C_DIM << 16)); // tile_dim0=100
  g1[4] = (int)((uint32_t)rows & 0xFFFFu);     // tile_dim1=rows, tile_dim2=0
  g1[5] = (int)stride0;                        // tensor_dim0_stride[31:0]
  g1[6] = 0;                                   // stride0[47:32], dim1_stride[15:0]
  g1[7] = 0;                                   // dim1_stride[47:16]

  v4i z4 = {0, 0, 0, 0};
#if defined(__clang_major__) && (__clang_major__ >= 23)
  v8i z8 = {0, 0, 0, 0, 0, 0, 0, 0};
  __builtin_amdgcn_tensor_load_to_lds(g0, g1, z4, z4, z8, 0);
#else
  __builtin_amdgcn_tensor_load_to_lds(g0, g1, z4, z4, 0);
#endif
}

// ---------------------------------------------------------------------------
// Main pass: TDM double-buffered streaming over row tiles; per-row wave32
// max/argmax + sum-exp shuffle reductions; binned accumulation in LDS; block
// flush to global workspace (45 floats: counts | conf_sum | acc_sum).
// ---------------------------------------------------------------------------
__global__ __launch_bounds__(BLOCK_THREADS)
void ece_partial_kernel(const float* __restrict__ logits,
                        const int* __restrict__ labels,
                        float* __restrict__ ws,
                        int N, int cntBase, int cntRem) {
  __shared__ __align__(16) float buf[2][TILE_ROWS * C_DIM];
  __shared__ float hist[3 * N_BINS];

  const int tid  = (int)threadIdx.x;
  const int lane = tid & 31;
  const int wave = tid >> 5;

  if (tid < 3 * N_BINS) hist[tid] = 0.0f;
  __syncthreads();

  // tiles for this block: blockIdx.x, +gridDim.x, ... (no runtime division)
  const int count = cntBase + (((int)blockIdx.x < cntRem) ? 1 : 0);

  const uint32_t lds_b0 = (uint32_t)(uintptr_t)&buf[0][0];
  const uint32_t lds_b1 = (uint32_t)(uintptr_t)&buf[1][0];

  // Prologue: wave 0 launches the first tile's DMA.
  if (count > 0 && wave == 0) {
    int rb   = (int)blockIdx.x * TILE_ROWS;
    int rows = min(TILE_ROWS, N - rb);
    tdm_load_tile_to_lds(lds_b0, logits + (size_t)rb * C_DIM, rows);
  }

  for (int i = 0; i < count; ++i) {
    const int tile    = (int)blockIdx.x + i * (int)gridDim.x;
    const int rowbase = tile * TILE_ROWS;
    const int rows    = min(TILE_ROWS, N - rowbase);
    const float* tb   = &buf[i & 1][0];

    if (wave == 0) {
      if (i + 1 < count) {   // prefetch next tile into the other buffer
        int nt    = (int)blockIdx.x + (i + 1) * (int)gridDim.x;
        int nrb   = nt * TILE_ROWS;
        int nrows = min(TILE_ROWS, N - nrb);
        tdm_load_tile_to_lds(((i + 1) & 1) ? lds_b1 : lds_b0,
                             logits + (size_t)nrb * C_DIM, nrows);
        __builtin_amdgcn_s_wait_tensorcnt(1);   // tile i landed (in-order)
      } else {
        __builtin_amdgcn_s_wait_tensorcnt(0);
      }
    }
    __syncthreads();
    asm volatile("" ::: "memory");  // TDM wrote LDS behind the compiler's back

    for (int r = wave; r < rows; r += WAVES_PER_BLK) {
      // 100 elems = 25 lanes x float4 (row stride 400 B = 25*16 B -> aligned)
      const bool active = (lane < LANES_USED);
      const float4* rp4 =
          reinterpret_cast<const float4*>(tb + r * C_DIM);
      float4 e;
      if (active) e = rp4[lane];                 // single ds_load_b128
      else        e = make_float4(-3.0e38f, -3.0e38f, -3.0e38f, -3.0e38f);

      // in-lane max/argmax, first occurrence wins (strict >)
      float m = e.x; int mi = lane * 4;
      if (e.y > m) { m = e.y; mi = lane * 4 + 1; }
      if (e.z > m) { m = e.z; mi = lane * 4 + 2; }
      if (e.w > m) { m = e.w; mi = lane * 4 + 3; }
      if (!active) mi = 0x7FFFFFFF;

      // wave32 max+argmax (first-occurrence tie-break, matching jnp.argmax)
      #pragma unroll
      for (int off = 16; off > 0; off >>= 1) {
        float om = __shfl_xor(m, off, 32);
        int  omi = __shfl_xor(mi, off, 32);
        if (om > m || (om == m && omi < mi)) { m = om; mi = omi; }
      }

      float s = 0.0f;
      if (active)
        s = __expf(e.x - m) + __expf(e.y - m) + __expf(e.z - m) + __expf(e.w - m);
      #pragma unroll
      for (int off = 16; off > 0; off >>= 1) s += __shfl_xor(s, off, 32);

      if (lane == 0) {
        const int row    = rowbase + r;
        const float conf = 1.0f / s;  // exp(m-m)/Σexp = max softmax prob
        const float acc  = (mi == labels[row]) ? 1.0f : 0.0f;
        // searchsorted(linspace(0,1,16), conf, 'left') - 1 ; conf∈(1/C, 1]
        int hi = N_BINS;
        #pragma unroll
        for (int b = 1; b < N_BINS; ++b) {
          if ((float)b / (float)N_BINS >= conf) { hi = b; break; }
        }
        const int bin = hi - 1;
        atomicAdd(&hist[bin], 1.0f);
        atomicAdd(&hist[N_BINS + bin], conf);
        atomicAdd(&hist[2 * N_BINS + bin], acc);
      }
    }
    __syncthreads();  // all waves done with buf[i&1] before it is overwritten
  }

  if (tid < 3 * N_BINS) atomicAdd(&ws[tid], hist[tid]);
}

__global__ void ece_zero_ws(float* ws) {
  if (threadIdx.x < 3 * N_BINS) ws[threadIdx.x] = 0.0f;
}

__global__ void ece_finalize(const float* __restrict__ ws,
                             float* __restrict__ out, float n) {
  if (threadIdx.x == 0) {
    float ece = 0.0f;
    for (int b = 0; b < N_BINS; ++b) {
      float cnt = ws[b];
      if (cnt > 0.0f) {
        float denom    = fmaxf(cnt, 1.0f);
        float avg_conf = ws[N_BINS + b] / denom;
        float avg_acc  = ws[2 * N_BINS + b] / denom;
        ece += fabsf(avg_conf - avg_acc) * (cnt / n);
      }
    }
    out[0] = ece;
  }
}

extern "C" void kernel_launch(void* const* d_in, const int* in_sizes, int n_in,
                              void* d_out, int out_size, void* d_ws, size_t ws_size,
                              hipStream_t stream) {
  (void)n_in; (void)out_size; (void)ws_size;
  const float* logits = (const float*)d_in[0];
  const int*   labels = (const int*)d_in[1];
  float* ws  = (float*)d_ws;
  float* out = (float*)d_out;

  const int N     = in_sizes[1];                       // labels count = rows
  const int tiles = (N + TILE_ROWS - 1) / TILE_ROWS;
  const int grid  = tiles < 2048 ? tiles : 2048;
  const int cntBase = tiles / grid;                    // host-side division
  const int cntRem  = tiles % grid;

  ece_zero_ws<<<1, 64, 0, stream>>>(ws);
  ece_partial_kernel<<<grid, BLOCK_THREADS, 0, stream>>>(logits, labels, ws,
                                                         N, cntBase, cntRem);
  ece_finalize<<<1, 32, 0, stream>>>(ws, out, (float)N);
}